// ButterworthFilter_24816321037021
// MI455X (gfx1250) — compile-verified
//
#include <hip/hip_runtime.h>

// CDNA5 / gfx1250 block-state-space IIR (sosfilt) using V_WMMA_F32_16X16X4_F32.
//
// x: [B=32, C=64, T=4096] f32, sos: [4,6] f32. 2048 channels, 16 per wave.
// Per 16-step time block (exact LTI block formula, 8-state cascade):
//   Y  = T16 * X + H * S      (T16: 16x16 Toeplitz impulse response, H: 16x8)
//   S' = F   * X + A * S      (F: 8x16, A: 8x8)
// -> 12 chained v_wmma_f32_16x16x4_f32 per 16 samples x 16 channels.
//
// State handoff trick: rows of [F;A] are permuted (rho) so that the S' GEMM's
// D-operand places each state component in exactly the lane+VGPR where the
// next iteration's B-operand needs it. The recurrence is carried purely in
// registers: sb0=(cs[0],cs[1]), sb1=(cs[2],cs[3]) for BOTH wave halves.
// No LDS, no shuffles, no barriers on the serial path.

typedef __attribute__((ext_vector_type(2))) float v2f;
typedef __attribute__((ext_vector_type(4))) float v4f;
typedef __attribute__((ext_vector_type(8))) float v8f;

#define T_LEN 4096
#define NCH   2048          // 32 * 64
#define BLK   16
#define NBLK  (T_LEN / BLK) // 256

// ws layout (floats):
//   [   0..255 ] Tm[m][k] : y response to impulse at k          (16x16)
//   [ 256..511 ] Hm[m][k] : y response to unit state k (k<8)    (16x16, cols 8..15 = 0)
//   [ 512..767 ] Fm[rho(r)][k] : final state r from impulse k   (16x16, other rows 0)
//   [ 768..1023] Am[rho(r)][k] : final state r from unit state k(16x16, other rows 0)
//
// rho maps state r -> D-matrix row so the S' GEMM output lands where the next
// B-operand slot lives:
//   B slot K=j is in lanes 0-15  for j in {0,1,4,5} -> D rows {0,1,2,3}
//   B slot K=j is in lanes 16-31 for j in {2,3,6,7} -> D rows {8,9,10,11}

__global__ void iir_setup_kernel(const float* __restrict__ sos, float* __restrict__ ws)
{
    const int lane = threadIdx.x;
    for (int i = lane; i < 1024; i += 32) ws[i] = 0.0f;
    __syncthreads();

    if (lane < 24) {
        const int rho[8] = {0, 1, 8, 9, 2, 3, 10, 11};

        float b0[4], b1[4], b2[4], a1[4], a2[4];
        #pragma unroll
        for (int s = 0; s < 4; ++s) {
            b0[s] = sos[s * 6 + 0];
            b1[s] = sos[s * 6 + 1];
            b2[s] = sos[s * 6 + 2];
            a1[s] = sos[s * 6 + 4];
            a2[s] = sos[s * 6 + 5];
        }
        float z1[4] = {0.f, 0.f, 0.f, 0.f};
        float z2[4] = {0.f, 0.f, 0.f, 0.f};

        const bool imp = (lane < 16);
        int col;
        if (imp) {
            col = lane;                         // impulse at time index 'col'
        } else {
            col = lane - 16;                    // unit initial state component 'col'
            const int sec = col >> 1;
            if (col & 1) z2[sec] = 1.0f; else z1[sec] = 1.0f;
        }
        float* Ym = imp ? (ws + 0)   : (ws + 256);
        float* Sm = imp ? (ws + 512) : (ws + 768);

        for (int n = 0; n < 16; ++n) {
            float w = (imp && (n == col)) ? 1.0f : 0.0f;
            #pragma unroll
            for (int s = 0; s < 4; ++s) {
                float y = fmaf(b0[s], w, z1[s]);
                z1[s]   = fmaf(b1[s], w, fmaf(-a1[s], y, z2[s]));
                z2[s]   = fmaf(b2[s], w, -a2[s] * y);
                w = y;
            }
            Ym[n * 16 + col] = w;               // column 'col' of T16 / H
        }
        #pragma unroll
        for (int s = 0; s < 4; ++s) {
            Sm[rho[2 * s + 0] * 16 + col] = z1[s]; // permuted column 'col' of F / A
            Sm[rho[2 * s + 1] * 16 + col] = z2[s];
        }
    }
}

__global__ void __launch_bounds__(32)
iir_block_kernel(const float* __restrict__ x, const float* __restrict__ ws,
                 float* __restrict__ out)
{
    const int lane = threadIdx.x;
    const int m    = lane & 15;   // A-operand row / channel-in-group / D-operand column
    const int half = lane >> 4;   // selects K pair for A/B operands, M half for D

    const int ch = blockIdx.x * 16 + m;
    const float* __restrict__ xc = x   + (size_t)ch * T_LEN;
    float*       __restrict__ oc = out + (size_t)ch * T_LEN;

    // A operands (16x4 f32 layout: lanes 0-15 -> K={4q,4q+1}, lanes 16-31 -> {4q+2,4q+3})
    v2f Ta[4], Fa[4], Ha[2], Aa[2];
    #pragma unroll
    for (int q = 0; q < 4; ++q) {
        const int k0 = 4 * q + 2 * half;
        Ta[q] = (v2f){ws[      m * 16 + k0], ws[      m * 16 + k0 + 1]};
        Fa[q] = (v2f){ws[512 + m * 16 + k0], ws[512 + m * 16 + k0 + 1]};
    }
    #pragma unroll
    for (int q = 0; q < 2; ++q) {
        const int k0 = 4 * q + 2 * half;
        Ha[q] = (v2f){ws[256 + m * 16 + k0], ws[256 + m * 16 + k0 + 1]};
        Aa[q] = (v2f){ws[768 + m * 16 + k0], ws[768 + m * 16 + k0 + 1]};
    }

    // register-carried state B operands (zero initial conditions)
    v2f sb0 = (v2f){0.f, 0.f};   // K rows {2h, 2h+1}
    v2f sb1 = (v2f){0.f, 0.f};   // K rows {4+2h, 4+2h+1}

    for (int blk = 0; blk < NBLK; ++blk) {
        const int t0 = blk * BLK;

        // X tile as B operand: lane holds N=channel m, K rows 4q + 2*half + {0,1}
        v2f xb[4];
        #pragma unroll
        for (int q = 0; q < 4; ++q) {
            xb[q] = *(const v2f*)(xc + t0 + 4 * q + 2 * half);  // global_load_b64
        }
        if (blk + 1 < NBLK) __builtin_prefetch(xc + t0 + BLK, 0, 3);

        v8f cy = {};  // Y accumulator  (16x16)
        v8f cs = {};  // permuted S' accumulator
        // state-independent chains first (pipeline freely across iterations)
        #pragma unroll
        for (int q = 0; q < 4; ++q) {
            cy = __builtin_amdgcn_wmma_f32_16x16x4_f32(false, Ta[q], false, xb[q],
                                                       (short)0, cy, false, false);
            cs = __builtin_amdgcn_wmma_f32_16x16x4_f32(false, Fa[q], false, xb[q],
                                                       (short)0, cs, false, false);
        }
        // state-dependent tail: the only loop-carried dependence (2 WMMAs deep)
        cy = __builtin_amdgcn_wmma_f32_16x16x4_f32(false, Ha[0], false, sb0,
                                                   (short)0, cy, false, false);
        cs = __builtin_amdgcn_wmma_f32_16x16x4_f32(false, Aa[0], false, sb0,
                                                   (short)0, cs, false, false);
        cy = __builtin_amdgcn_wmma_f32_16x16x4_f32(false, Ha[1], false, sb1,
                                                   (short)0, cy, false, false);
        cs = __builtin_amdgcn_wmma_f32_16x16x4_f32(false, Aa[1], false, sb1,
                                                   (short)0, cs, false, false);

        // D layout: lane holds channel m, time rows v + 8*half -> 2x b128 NT stores
        v4f lo = (v4f){cy[0], cy[1], cy[2], cy[3]};
        v4f hi = (v4f){cy[4], cy[5], cy[6], cy[7]};
        __builtin_nontemporal_store(lo, (v4f*)(oc + t0 + 8 * half));
        __builtin_nontemporal_store(hi, (v4f*)(oc + t0 + 8 * half + 4));

        // rho-permuted S' lands exactly where next iteration's B operand needs it
        sb0 = (v2f){cs[0], cs[1]};
        sb1 = (v2f){cs[2], cs[3]};
    }
}

extern "C" void kernel_launch(void* const* d_in, const int* in_sizes, int n_in,
                              void* d_out, int out_size, void* d_ws, size_t ws_size,
                              hipStream_t stream) {
    const float* x   = (const float*)d_in[0];   // [32,64,4096] f32
    const float* sos = (const float*)d_in[1];   // [4,6] f32
    float* out = (float*)d_out;
    float* ws  = (float*)d_ws;                  // needs 4 KiB

    iir_setup_kernel<<<1, 32, 0, stream>>>(sos, ws);
    iir_block_kernel<<<NCH / 16, 32, 0, stream>>>(x, ws, out);
}